// StackedBranchNet_3401614098663
// MI455X (gfx1250) — compile-verified
//
#include <hip/hip_runtime.h>
#include <math.h>

typedef __attribute__((ext_vector_type(2))) float v2f;
typedef __attribute__((ext_vector_type(8))) float v8f;

#define INPUT_DIM   100
#define LATENT_DIM  128
#define HIDDEN_DIM  64
#define BATCH       16384

#define ROWS_PER_BLOCK 64     // 2 waves x two 16-row batch tiles each
#define WAVES          2
#define BLOCK          (WAVES * 32)

// Hardware tanh if available (V_TANH_F32 is a CDNA5 TRANS op), else OCML.
#if __has_builtin(__builtin_amdgcn_tanhf)
  #define FAST_TANH(x) __builtin_amdgcn_tanhf(x)
#elif __has_builtin(__builtin_amdgcn_tanh_f32)
  #define FAST_TANH(x) __builtin_amdgcn_tanh_f32(x)
#else
  #define FAST_TANH(x) tanhf(x)
#endif

__global__ __launch_bounds__(BLOCK)
void branchnet_wmma_f32(const float* __restrict__ u,
                        const float* __restrict__ W1,
                        const float* __restrict__ b1,
                        const float* __restrict__ W2,
                        const float* __restrict__ b2,
                        float* __restrict__ out)
{
    // u tile (64x100) and W1[k] slice (64x100) are contiguous in global
    // memory -> coalesced float4 staging into LDS.
    __shared__ alignas(16) float sU[ROWS_PER_BLOCK * INPUT_DIM];   // 25.6 KB
    __shared__ alignas(16) float sB[HIDDEN_DIM * INPUT_DIM];       // 25.6 KB
    __shared__ float sb1[HIDDEN_DIM];
    __shared__ float sW2[HIDDEN_DIM];
    __shared__ float sRed[WAVES][32];

    const int tid = threadIdx.x;
    const int k   = blockIdx.y;                 // latent index
    const int b0  = blockIdx.x * ROWS_PER_BLOCK;

    // ---- cooperative staging (contiguous, vectorized) ----
    {
        const float4* uSrc = (const float4*)(u  + (size_t)b0 * INPUT_DIM);
        const float4* wSrc = (const float4*)(W1 + (size_t)k * (HIDDEN_DIM * INPUT_DIM));
        float4* uDst = (float4*)sU;
        float4* wDst = (float4*)sB;
        const int NV = ROWS_PER_BLOCK * INPUT_DIM / 4;   // 1600 float4 each
        for (int i = tid; i < NV; i += BLOCK) {
            uDst[i] = uSrc[i];
            wDst[i] = wSrc[i];
        }
        if (tid < HIDDEN_DIM) {
            sb1[tid] = b1[k * HIDDEN_DIM + tid];
            sW2[tid] = W2[k * HIDDEN_DIM + tid];
        }
    }
    __syncthreads();

    const int wave = tid >> 5;     // 0..1
    const int lane = tid & 31;
    const int ln   = lane & 15;    // 0..15
    const int hs   = lane >> 4;    // 0/1 half-wave selector

    // ---- grouped GEMM: 25 K-steps of V_WMMA_F32_16X16X4_F32 ----
    // Each wave owns TWO 16-row batch tiles so every B fragment feeds 2 WMMAs.
    // A (16x4): lane<16 row=ln K={0,1}; lane>=16 row=ln K={2,3}  -> float2
    // B (4x16): half-wave selects K pair {hs*2,hs*2+1}, N=ln     -> float2
    v8f acc0[4], acc1[4];
#pragma unroll
    for (int t = 0; t < 4; ++t) {
        acc0[t] = (v8f){0.f,0.f,0.f,0.f,0.f,0.f,0.f,0.f};
        acc1[t] = (v8f){0.f,0.f,0.f,0.f,0.f,0.f,0.f,0.f};
    }

    const float* aBase0 = sU + (wave * 32 + ln) * INPUT_DIM + hs * 2;        // tile 0
    const float* aBase1 = aBase0 + 16 * INPUT_DIM;                           // tile 1
    const float* bBase  = sB + ln * INPUT_DIM + hs * 2;

#pragma unroll
    for (int s = 0; s < 25; ++s) {
        const v2f a0 = *(const v2f*)(aBase0 + 4 * s);
        const v2f a1 = *(const v2f*)(aBase1 + 4 * s);
#pragma unroll
        for (int t = 0; t < 4; ++t) {
            const v2f b = *(const v2f*)(bBase + t * 16 * INPUT_DIM + 4 * s);
            acc0[t] = __builtin_amdgcn_wmma_f32_16x16x4_f32(
                false, a0, false, b, (short)0, acc0[t], false, false);
            acc1[t] = __builtin_amdgcn_wmma_f32_16x16x4_f32(
                false, a1, false, b, (short)0, acc1[t], false, false);
        }
    }

    // ---- epilogue: tanh (hardware trans op), scale by W2, reduce over h ----
    // C layout: VGPR r, lanes 0-15 -> row M=r, lanes 16-31 -> row M=8+r; N=ln.
#pragma unroll
    for (int j = 0; j < 2; ++j) {
        float p[8];
#pragma unroll
        for (int r = 0; r < 8; ++r) p[r] = 0.f;

#pragma unroll
        for (int t = 0; t < 4; ++t) {
            const int h  = t * 16 + ln;
            const float bb = sb1[h];
            const float ww = sW2[h];
#pragma unroll
            for (int r = 0; r < 8; ++r) {
                const float c  = (j == 0) ? acc0[t][r] : acc1[t][r];
                const float hv = FAST_TANH(c + bb);
                p[r] += hv * ww;
            }
        }

        // butterfly reduction over the 16 lanes of each half-wave
#pragma unroll
        for (int r = 0; r < 8; ++r) {
#pragma unroll
            for (int off = 8; off >= 1; off >>= 1)
                p[r] += __shfl_xor(p[r], off, 32);
        }

        if (ln == 0) {
#pragma unroll
            for (int r = 0; r < 8; ++r)
                sRed[wave][j * 16 + hs * 8 + r] = p[r];
        }
    }
    __syncthreads();

    // 64 output rows per block: out[b0+tid][k]
    if (tid < ROWS_PER_BLOCK) {
        const int w = tid >> 5;       // owning wave (32 rows each)
        const int m = tid & 31;       // row within wave's span
        out[(size_t)(b0 + tid) * LATENT_DIM + k] = sRed[w][m] + b2[k];
    }
}

extern "C" void kernel_launch(void* const* d_in, const int* in_sizes, int n_in,
                              void* d_out, int out_size, void* d_ws, size_t ws_size,
                              hipStream_t stream) {
    const float* u  = (const float*)d_in[0];
    const float* W1 = (const float*)d_in[1];
    const float* b1 = (const float*)d_in[2];
    const float* W2 = (const float*)d_in[3];
    const float* b2 = (const float*)d_in[4];
    float* out = (float*)d_out;

    dim3 grid(BATCH / ROWS_PER_BLOCK, LATENT_DIM);  // 256 x 128 blocks
    branchnet_wmma_f32<<<grid, BLOCK, 0, stream>>>(u, W1, b1, W2, b2, out);
}